// OuterProductMean_89455578841546
// MI455X (gfx1250) — compile-verified
//
#include <hip/hip_runtime.h>

#define S_DEPTH 128
#define N_RES   512
#define E_DIM   256
#define P_DIM   32
#define C_DIM   128

typedef __attribute__((ext_vector_type(2))) float v2f;
typedef __attribute__((ext_vector_type(8))) float v8f;

__device__ __forceinline__ v8f wmma_f32_k4(v2f a, v2f b, v8f c) {
  // D = A(16x4, f32) * B(4x16, f32) + C(16x16, f32)
  return __builtin_amdgcn_wmma_f32_16x16x4_f32(
      false, a, false, b, (short)0, c, false, false);
}

// ---------------------------------------------------------------------------
// Kernel 1: LayerNorm over E + dual projection E->P.
// One wave per (s,n) row; results transposed through LDS so a/b are stored
// PLANAR [p][s*N+n] -> all stage-2 loads become lane-contiguous.
// ---------------------------------------------------------------------------
__global__ __launch_bounds__(256) void opm_ln_proj(
    const float* __restrict__ msa, const float* __restrict__ gamma,
    const float* __restrict__ beta, const float* __restrict__ Wa,
    const float* __restrict__ ba, const float* __restrict__ Wb,
    const float* __restrict__ bb, float* __restrict__ a_out,
    float* __restrict__ b_out) {
  __shared__ float xs[8 * E_DIM];      // normalized rows (8 KB)
  __shared__ float als[P_DIM * 8];     // [p][wave] transpose buffers (1 KB)
  __shared__ float bls[P_DIM * 8];
  const int wave = threadIdx.x >> 5;
  const int lane = threadIdx.x & 31;
  const int row  = blockIdx.x * 8 + wave;  // flattened (s*N + n), 8 rows/block
  const float* xr = msa + (size_t)row * E_DIM;

  float x[8];
  float s = 0.f, s2 = 0.f;
#pragma unroll
  for (int i = 0; i < 8; ++i) {
    x[i] = xr[lane * 8 + i];
    s  += x[i];
    s2 += x[i] * x[i];
  }
#pragma unroll
  for (int off = 16; off >= 1; off >>= 1) {  // wave32 butterfly reduction
    s  += __shfl_xor(s,  off, 32);
    s2 += __shfl_xor(s2, off, 32);
  }
  const float mu  = s * (1.0f / E_DIM);
  const float var = s2 * (1.0f / E_DIM) - mu * mu;
  const float inv = rsqrtf(var + 1e-5f);
#pragma unroll
  for (int i = 0; i < 8; ++i) {
    const int e = lane * 8 + i;
    xs[wave * E_DIM + e] = (x[i] - mu) * inv * gamma[e] + beta[e];
  }
  __syncthreads();

  const int p = lane;  // P == 32 == wave width
  float aa = 0.f, ab = 0.f;
  for (int k = 0; k < E_DIM; ++k) {
    const float xk = xs[wave * E_DIM + k];  // LDS broadcast, conflict-free
    aa = fmaf(xk, Wa[k * P_DIM + p], aa);
    ab = fmaf(xk, Wb[k * P_DIM + p], ab);
  }
  als[lane * 8 + wave] = aa + ba[p];
  bls[lane * 8 + wave] = ab + bb[p];
  __syncthreads();

  // Planar store: thread t handles plane tp, row offset tj (32 B runs).
  const int tp = threadIdx.x >> 3;
  const int tj = threadIdx.x & 7;
  const size_t SN = (size_t)S_DEPTH * N_RES;
  const size_t rowbase = (size_t)blockIdx.x * 8;
  a_out[(size_t)tp * SN + rowbase + tj] = als[tp * 8 + tj];
  b_out[(size_t)tp * SN + rowbase + tj] = bls[tp * 8 + tj];
}

// ---------------------------------------------------------------------------
// Kernel 2: outer[p][n][m] = (1/S) * sum_s a_pl[p][s][n] * b_pl[p][s][m]
// Batched GEMM over p (K = S = 128). Per wave: 64x64 tile = 4x4 WMMA tiles
// -> 16 operand loads per 16 WMMAs. All global accesses lane-contiguous.
// ---------------------------------------------------------------------------
__global__ __launch_bounds__(256) void opm_outer(
    const float* __restrict__ a, const float* __restrict__ b,
    float* __restrict__ outer) {
  const int wave = threadIdx.x >> 5;
  const int lane = threadIdx.x & 31;
  const int w    = blockIdx.x * 8 + wave;  // 2048 waves total
  const int p    = w >> 6;                 // batch index, 0..31
  const int rem  = w & 63;                 // 8x8 grid of 64x64 tiles
  const int tn   = (rem >> 3) << 6;
  const int tm   = (rem & 7) << 6;

  const int half = lane >> 4;   // A/B K striping: 0 -> k0,k0+1 ; 1 -> k0+2,k0+3
  const int r    = lane & 15;
  const size_t SN = (size_t)S_DEPTH * N_RES;
  const size_t NN = (size_t)N_RES * N_RES;
  const float* apl = a + (size_t)p * SN;
  const float* bpl = b + (size_t)p * SN;

  v8f acc[4][4] = {};
  for (int s0 = 0; s0 < S_DEPTH; s0 += 4) {
    const int kl = s0 + (half << 1);
    v2f Ar[4], Br[4];
#pragma unroll
    for (int t = 0; t < 4; ++t) {
      const size_t a0 = (size_t)kl * N_RES + (tn + t * 16 + r);
      Ar[t].x = apl[a0];
      Ar[t].y = apl[a0 + N_RES];
      const size_t b0 = (size_t)kl * N_RES + (tm + t * 16 + r);
      Br[t].x = bpl[b0];
      Br[t].y = bpl[b0 + N_RES];
    }
    if (s0 + 4 < S_DEPTH) {  // global_prefetch_b8 for next K-chunk
      __builtin_prefetch(apl + (size_t)(s0 + 4) * N_RES + tn + r, 0, 0);
      __builtin_prefetch(bpl + (size_t)(s0 + 4) * N_RES + tm + r, 0, 0);
    }
#pragma unroll
    for (int ni = 0; ni < 4; ++ni)
#pragma unroll
      for (int mi = 0; mi < 4; ++mi)
        acc[ni][mi] = wmma_f32_k4(Ar[ni], Br[mi], acc[ni][mi]);
  }

  const float scale = 1.0f / (float)S_DEPTH;
  float* op = outer + (size_t)p * NN;
#pragma unroll
  for (int ni = 0; ni < 4; ++ni)
#pragma unroll
    for (int mi = 0; mi < 4; ++mi)
#pragma unroll
      for (int v = 0; v < 8; ++v) {
        const int n = tn + ni * 16 + half * 8 + v;  // D row = VGPR + 8*(lane>=16)
        const int m = tm + mi * 16 + r;             // D col = lane%16
        op[(size_t)n * N_RES + m] = acc[ni][mi][v] * scale;  // coalesced
      }
}

// ---------------------------------------------------------------------------
// Kernel 3: out[nm, c] = outer_pl[:, nm] @ Wp + bp   ([262144,32] x [32,128])
// Per wave: 32-row strip x 128 cols = 2x8 WMMA acc tiles; B shared by both
// row tiles. outer is planar [p][nm] -> A loads lane-contiguous.
// ---------------------------------------------------------------------------
__global__ __launch_bounds__(256) void opm_pair(
    const float* __restrict__ outer, const float* __restrict__ Wp,
    const float* __restrict__ bp, float* __restrict__ out) {
  const int wave = threadIdx.x >> 5;
  const int lane = threadIdx.x & 31;
  const int w    = blockIdx.x * 8 + wave;  // 8192 waves
  const size_t rows0 = (size_t)w * 32;     // flattened (n*N + m) row base
  const int half = lane >> 4;
  const int r    = lane & 15;
  const size_t NN = (size_t)N_RES * N_RES;

  v8f acc[2][8];
#pragma unroll
  for (int ct = 0; ct < 8; ++ct) {
    const float bias = bp[ct * 16 + r];    // D col = lane%16 for every VGPR
#pragma unroll
    for (int v = 0; v < 8; ++v) {
      acc[0][ct][v] = bias;
      acc[1][ct][v] = bias;
    }
  }

  for (int k0 = 0; k0 < P_DIM; k0 += 4) {
    const int kl = k0 + (half << 1);
    v2f Ar[2];
#pragma unroll
    for (int rt = 0; rt < 2; ++rt) {
      Ar[rt].x = outer[(size_t)(kl + 0) * NN + rows0 + rt * 16 + r];
      Ar[rt].y = outer[(size_t)(kl + 1) * NN + rows0 + rt * 16 + r];
    }
#pragma unroll
    for (int ct = 0; ct < 8; ++ct) {
      v2f Br;
      Br.x = Wp[(size_t)(kl + 0) * C_DIM + ct * 16 + r];
      Br.y = Wp[(size_t)(kl + 1) * C_DIM + ct * 16 + r];
      acc[0][ct] = wmma_f32_k4(Ar[0], Br, acc[0][ct]);
      acc[1][ct] = wmma_f32_k4(Ar[1], Br, acc[1][ct]);
    }
  }

#pragma unroll
  for (int rt = 0; rt < 2; ++rt)
#pragma unroll
    for (int ct = 0; ct < 8; ++ct)
#pragma unroll
      for (int v = 0; v < 8; ++v)
        out[(rows0 + rt * 16 + half * 8 + v) * C_DIM + ct * 16 + r] =
            acc[rt][ct][v];
}

// ---------------------------------------------------------------------------
extern "C" void kernel_launch(void* const* d_in, const int* in_sizes, int n_in,
                              void* d_out, int out_size, void* d_ws,
                              size_t ws_size, hipStream_t stream) {
  const float* msa   = (const float*)d_in[0];
  const float* gamma = (const float*)d_in[1];
  const float* beta  = (const float*)d_in[2];
  const float* Wa    = (const float*)d_in[3];
  const float* ba    = (const float*)d_in[4];
  const float* Wb    = (const float*)d_in[5];
  const float* bb    = (const float*)d_in[6];
  const float* Wp    = (const float*)d_in[7];
  const float* bp    = (const float*)d_in[8];
  float* out = (float*)d_out;

  // Workspace: a_pl[P][S*N] | b_pl[P][S*N] | outer_pl[P][N*N]  (~50.3 MB)
  float* a_ws     = (float*)d_ws;
  float* b_ws     = a_ws + (size_t)P_DIM * S_DEPTH * N_RES;
  float* outer_ws = b_ws + (size_t)P_DIM * S_DEPTH * N_RES;

  // 1) LayerNorm + projections: 65536 rows, 1 wave each, 8 waves/block.
  opm_ln_proj<<<(S_DEPTH * N_RES) / 8, 256, 0, stream>>>(
      msa, gamma, beta, Wa, ba, Wb, bb, a_ws, b_ws);

  // 2) Batched outer-product GEMM: 32 p-batches x (8x8) 64x64 tiles.
  opm_outer<<<(32 * 8 * 8) / 8, 256, 0, stream>>>(a_ws, b_ws, outer_ws);

  // 3) Pair projection: 262144/32 row-strips, 8 waves/block.
  opm_pair<<<((N_RES * N_RES) / 32) / 8, 256, 0, stream>>>(
      outer_ws, Wp, bp, out);
}